// InvariantPointAttention_19344532701693
// MI455X (gfx1250) — compile-verified
//
#include <hip/hip_runtime.h>
#include <hip/hip_bf16.h>

// Problem constants (match reference)
#define Nn   512
#define CSd  384
#define CZd  128
#define Hd   12
#define CHd  16
#define PQd  4
#define PVd  8
#define CATd 2112   // H*(CH + PV*4 + CZ) = 12*176

typedef __attribute__((ext_vector_type(2))) float v2f;
typedef __attribute__((ext_vector_type(8))) float v8f;

// D = A(16x4) * B(4x16) + C, fp32 matrix pipe (V_WMMA_F32_16X16X4_F32).
__device__ __forceinline__ v8f wmma4(v2f a, v2f b, v8f c) {
  return __builtin_amdgcn_wmma_f32_16x16x4_f32(false, a, false, b, (short)0, c,
                                               false, false);
}

// ---------------------------------------------------------------------------
// Generic row-major GEMM: C[M x N] = A[M x K] * B[K x N] + bias[N].
// One wave32 computes one 16x16 tile with K-steps of 4 (f32 WMMA).
// Shapes are compile-time so all addressing folds to immediate offsets.
// Ragged N handled by clamping the B column address (garbage feeds only
// unstored D columns per WMMA semantics) and guarding the store.
// ---------------------------------------------------------------------------
template <int N, int K, int LDA, int LDB, int LDC>
__launch_bounds__(256)
__global__ void gemm_bias_kernel(const float* __restrict__ A,
                                 const float* __restrict__ B,
                                 const float* __restrict__ bias,
                                 float* __restrict__ C, int M) {
  constexpr int tilesN = (N + 15) >> 4;
  const int lane = threadIdx.x & 31;
  const int wave = threadIdx.x >> 5;
  int tile = blockIdx.x * 8 + wave;
  if (tile >= (M >> 4) * tilesN) return;
  const int tm = tile / tilesN, tn = tile % tilesN;
  const int m0 = tm << 4, n0 = tn << 4;
  const int row = lane & 15;          // A: M sub-index / B: N sub-index
  const int kk = (lane >> 4) << 1;    // K sub-offset {0,2}
  const int ncol = n0 + row;
  const int colc = (ncol < N) ? ncol : (N - 1);   // clamp, no predication

  const float* Ap = A + (long)(m0 + row) * LDA + kk;
  const float* Bp = B + (long)kk * LDB + colc;
  v8f acc = {};
#pragma unroll 8
  for (int k0 = 0; k0 < K; k0 += 4) {
    v2f a, b;
    a.x = Ap[k0];
    a.y = Ap[k0 + 1];
    b.x = Bp[(long)k0 * LDB];
    b.y = Bp[(long)k0 * LDB + LDB];
    acc = wmma4(a, b, acc);
  }
  if (ncol >= N) return;
  const float bv = bias ? bias[ncol] : 0.0f;
  alignas(32) float accA[8];
  *(v8f*)accA = acc;
  const int mBase = m0 + ((lane >> 4) << 3);
#pragma unroll
  for (int g = 0; g < 8; ++g)
    C[(long)(mBase + g) * LDC + ncol] = accA[g] + bv;
}

// ---------------------------------------------------------------------------
// Point packing: rotate+translate projected points into the global frame.
// qpraw: [N][3][H*PQ] -> qpts [N][H][12] (p*3+i), qn [N][H] = sum of squares
// ---------------------------------------------------------------------------
__global__ void pack_qpts_kernel(const float* __restrict__ qpraw,
                                 const float* __restrict__ rot,
                                 const float* __restrict__ trans,
                                 float* __restrict__ qpts,
                                 float* __restrict__ qn) {
  int t = blockIdx.x * blockDim.x + threadIdx.x;
  if (t >= Nn * Hd) return;
  int n = t / Hd, h = t % Hd;
  const float* R = rot + n * 9;
  const float* T = trans + n * 3;
  float s = 0.0f;
#pragma unroll
  for (int p = 0; p < PQd; ++p) {
    float lx = qpraw[n * 144 + 0 * 48 + h * PQd + p];
    float ly = qpraw[n * 144 + 1 * 48 + h * PQd + p];
    float lz = qpraw[n * 144 + 2 * 48 + h * PQd + p];
#pragma unroll
    for (int i = 0; i < 3; ++i) {
      float g = R[i * 3 + 0] * lx + R[i * 3 + 1] * ly + R[i * 3 + 2] * lz + T[i];
      qpts[n * 144 + h * 12 + p * 3 + i] = g;
      s += g * g;
    }
  }
  qn[n * Hd + h] = s;
}

// kvpraw: [N][3][H*12] -> kptsT [H][12][N] (k-points, transposed for B frags),
//                          vpts  [N][H][24], kn [N][H]
__global__ void pack_kvpts_kernel(const float* __restrict__ kvpraw,
                                  const float* __restrict__ rot,
                                  const float* __restrict__ trans,
                                  float* __restrict__ kptsT,
                                  float* __restrict__ vpts,
                                  float* __restrict__ kn) {
  int t = blockIdx.x * blockDim.x + threadIdx.x;
  if (t >= Nn * Hd) return;
  int n = t / Hd, h = t % Hd;
  const float* R = rot + n * 9;
  const float* T = trans + n * 3;
  float s = 0.0f;
#pragma unroll
  for (int p = 0; p < PQd + PVd; ++p) {
    float lx = kvpraw[n * 432 + 0 * 144 + h * 12 + p];
    float ly = kvpraw[n * 432 + 1 * 144 + h * 12 + p];
    float lz = kvpraw[n * 432 + 2 * 144 + h * 12 + p];
#pragma unroll
    for (int i = 0; i < 3; ++i) {
      float g = R[i * 3 + 0] * lx + R[i * 3 + 1] * ly + R[i * 3 + 2] * lz + T[i];
      if (p < PQd) {
        kptsT[(h * 12 + p * 3 + i) * Nn + n] = g;
        s += g * g;
      } else {
        vpts[n * 288 + h * 24 + (p - PQd) * 3 + i] = g;
      }
    }
  }
  kn[n * Hd + h] = s;
}

// kv [N][H*2CH] -> kT [H][CH][N]  (coalesced B fragments for the q*k^T WMMA)
__global__ void pack_kT_kernel(const float* __restrict__ kv,
                               float* __restrict__ kT) {
  int t = blockIdx.x * blockDim.x + threadIdx.x;
  if (t >= Nn * Hd * CHd) return;
  int n = t / (Hd * CHd);
  int rem = t % (Hd * CHd);
  int h = rem / CHd, c = rem % CHd;
  kT[(h * CHd + c) * Nn + n] = kv[n * 384 + h * 32 + c];
}

// ---------------------------------------------------------------------------
// Fused attention: one workgroup = (head h, 16-query block). 8 waves.
// Phase 1: logits via WMMA (q*k^T over K=16, q_pts*k_pts^T over K=12, scales
//          folded into A fragments) + bias + point-norm + mask -> LDS.
// Phase 2: softmax over 512 keys per row (wave32 shuffle reductions);
//          probabilities -> LDS and transposed aT[q][k][16] for o_pair pass.
// Phase 3: o = P*v and o_pt = P*v_pts via WMMA; 3 tasks x 2 split-K waves,
//          partials combined through LDS.
// ---------------------------------------------------------------------------
__launch_bounds__(256)
__global__ void attn_kernel(const float* __restrict__ q,
                            const float* __restrict__ kT,
                            const float* __restrict__ kv,
                            const float* __restrict__ qpts,
                            const float* __restrict__ kptsT,
                            const float* __restrict__ vpts,
                            const float* __restrict__ qn,
                            const float* __restrict__ kn,
                            const float* __restrict__ biasb,
                            const float* __restrict__ mask,
                            const float* __restrict__ head_w,
                            float* __restrict__ aT,
                            float* __restrict__ ocat,
                            float* __restrict__ opraw) {
  __shared__ float L[16][516];       // pad 512->516 to spread LDS banks
  __shared__ float Pp[3][16][16];    // split-K partials for phase 3

  const int h = blockIdx.x >> 5;          // 0..11
  const int qb = blockIdx.x & 31;         // 0..31
  const int q0 = qb << 4;
  const int lane = threadIdx.x & 31;
  const int wave = threadIdx.x >> 5;
  const int row = lane & 15;
  const int kk = (lane >> 4) << 1;

  const float hw = log1pf(__expf(head_w[h])) * 0.13608276348795434f;  // sqrt(1/54)
  const float s1 = 0.14433756729740643f;  // sqrt(1/(3*CH))
  const float s3 = 0.57735026918962584f;  // sqrt(1/3)

  // A fragments are invariant across key tiles: preload with scales folded in.
  v2f aq[4], ap[3];
  {
    const float* Qp = q + (long)(q0 + row) * 192 + h * CHd + kk;
#pragma unroll
    for (int s = 0; s < 4; ++s) { aq[s].x = Qp[4 * s] * s1; aq[s].y = Qp[4 * s + 1] * s1; }
    const float* Pq = qpts + (long)(q0 + row) * 144 + h * 12 + kk;
#pragma unroll
    for (int s = 0; s < 3; ++s) { ap[s].x = Pq[4 * s] * hw; ap[s].y = Pq[4 * s + 1] * hw; }
  }

  // ---- Phase 1: logits ----
  for (int kt = wave; kt < 32; kt += 8) {
    const int kcol = (kt << 4) + row;
    v8f acc = {};
    const float* Kp = kT + (long)(h * CHd + kk) * Nn + kcol;
#pragma unroll
    for (int s = 0; s < 4; ++s) {
      v2f b; b.x = Kp[(4 * s) * Nn]; b.y = Kp[(4 * s + 1) * Nn];
      acc = wmma4(aq[s], b, acc);
    }
    const float* KPp = kptsT + (long)(h * 12 + kk) * Nn + kcol;
#pragma unroll
    for (int s = 0; s < 3; ++s) {
      v2f b; b.x = KPp[(4 * s) * Nn]; b.y = KPp[(4 * s + 1) * Nn];
      acc = wmma4(ap[s], b, acc);
    }
    alignas(32) float accA[8];
    *(v8f*)accA = acc;
    const float knv = kn[kcol * Hd + h];
    const float mk = mask[kcol];
#pragma unroll
    for (int g = 0; g < 8; ++g) {
      const int m = g + ((lane >> 4) << 3);
      const int qi = q0 + m;
      float logit = accA[g]
                  + s3 * biasb[((long)qi * Nn + kcol) * Hd + h]
                  - 0.5f * hw * (qn[qi * Hd + h] + knv)
                  + (mask[qi] * mk - 1.0f) * 100000.0f;
      L[m][kcol] = logit;
    }
  }
  __syncthreads();

  // ---- Phase 2: softmax over keys (two rows per wave) ----
  for (int r = wave * 2; r < wave * 2 + 2; ++r) {
    float mx = -3.4e38f;
#pragma unroll
    for (int k2 = lane; k2 < Nn; k2 += 32) mx = fmaxf(mx, L[r][k2]);
#pragma unroll
    for (int off = 16; off; off >>= 1) mx = fmaxf(mx, __shfl_xor(mx, off, 32));
    float e[16];
    float sum = 0.0f;
    int idx = 0;
#pragma unroll
    for (int k2 = lane; k2 < Nn; k2 += 32) {
      float v = __expf(L[r][k2] - mx);
      e[idx++] = v;
      sum += v;
    }
#pragma unroll
    for (int off = 16; off; off >>= 1) sum += __shfl_xor(sum, off, 32);
    const float inv = 1.0f / sum;
    idx = 0;
#pragma unroll
    for (int k2 = lane; k2 < Nn; k2 += 32) {
      float p = e[idx++] * inv;
      L[r][k2] = p;
      aT[((long)(q0 + r) * Nn + k2) * 16 + h] = p;  // transposed for o_pair
    }
  }
  __syncthreads();

  // ---- Phase 3: o (task 0) and o_pt (tasks 1,2), 2 split-K waves per task ----
  v8f acc = {};
  const int task = wave >> 1;       // 0..2 for waves 0..5
  const int half = wave & 1;
  int pc = 0;
  if (wave < 6) {
    const int kbeg = half << 8;     // 0 or 256
    const float* Bp;
    long bstride;
    if (task == 0) {
      Bp = kv + (long)kbeg * 384 + h * 32 + 16 + row;   // v, element (k, c=row)
      bstride = 384;
    } else {
      pc = ((task - 1) << 4) + row;                     // 0..31; valid < 24
      const int pcl = (pc < 24) ? pc : 23;              // clamp address only
      Bp = vpts + (long)kbeg * 288 + h * 24 + pcl;
      bstride = 288;
    }
#pragma unroll 4
    for (int k0 = 0; k0 < 256; k0 += 4) {
      v2f a, b;
      a.x = L[row][kbeg + k0 + kk];
      a.y = L[row][kbeg + k0 + kk + 1];
      b.x = Bp[(k0 + kk) * bstride];
      b.y = Bp[(k0 + kk + 1) * bstride];
      acc = wmma4(a, b, acc);
    }
    if (half == 1) {
      alignas(32) float accA[8];
      *(v8f*)accA = acc;
#pragma unroll
      for (int g = 0; g < 8; ++g)
        Pp[task][g + ((lane >> 4) << 3)][row] = accA[g];
    }
  }
  __syncthreads();
  if (wave < 6 && half == 0) {
    alignas(32) float accA[8];
    *(v8f*)accA = acc;
    if (task == 0) {
#pragma unroll
      for (int g = 0; g < 8; ++g) {
        const int m = g + ((lane >> 4) << 3);
        ocat[(long)(q0 + m) * CATd + h * CHd + row] = accA[g] + Pp[0][m][row];
      }
    } else if (pc < 24) {
#pragma unroll
      for (int g = 0; g < 8; ++g) {
        const int m = g + ((lane >> 4) << 3);
        opraw[(long)(q0 + m) * 288 + h * 24 + pc] = accA[g] + Pp[task][m][row];
      }
    }
  }
}

// ---------------------------------------------------------------------------
// o_pair: per query qi, out[h][c] = sum_k P[h][k] * z[qi][k][c].
// (12x512)@(512x128) WMMA GEMM per workgroup; 8 waves = 8 column tiles.
// z is L2-resident after the b_bias pass -> this streams from L2.
// Padded A rows (h=12..15) read in-bounds workspace garbage that only feeds
// unstored D rows, so loads are unpredicated.
// ---------------------------------------------------------------------------
__launch_bounds__(256)
__global__ void opair_kernel(const float* __restrict__ aT,
                             const float* __restrict__ z,
                             float* __restrict__ ocat) {
  const int qi = blockIdx.x;
  const int lane = threadIdx.x & 31;
  const int wave = threadIdx.x >> 5;  // column tile 0..7
  const int row = lane & 15;
  const int kk = (lane >> 4) << 1;
  const int n0 = wave << 4;

  const float* Ap = aT + (long)qi * Nn * 16 + kk * 16 + row;  // [k][16]
  const float* Bp = z + (long)qi * Nn * CZd + (long)kk * CZd + n0 + row;

  v8f acc = {};
#pragma unroll 4
  for (int k0 = 0; k0 < Nn; k0 += 4) {
    if ((k0 & 28) == 0 && k0 + 64 < Nn)
      __builtin_prefetch(Bp + (long)(k0 + 64) * CZd + lane * 16, 0, 1);
    v2f a, b;
    a.x = Ap[k0 * 16];
    a.y = Ap[k0 * 16 + 16];
    b.x = Bp[(long)k0 * CZd];
    b.y = Bp[(long)k0 * CZd + CZd];
    acc = wmma4(a, b, acc);
  }
  alignas(32) float accA[8];
  *(v8f*)accA = acc;
#pragma unroll
  for (int g = 0; g < 8; ++g) {
    const int m = g + ((lane >> 4) << 3);  // head index
    if (m < Hd)
      ocat[(long)qi * CATd + 576 + m * CZd + n0 + row] = accA[g];
  }
}

// ---------------------------------------------------------------------------
// o_pt finalize: local = R^T (o_pt - t), norms, scatter into o_cat.
// ---------------------------------------------------------------------------
__global__ void finalize_opt_kernel(const float* __restrict__ opraw,
                                    const float* __restrict__ rot,
                                    const float* __restrict__ trans,
                                    float* __restrict__ ocat) {
  int t = blockIdx.x * blockDim.x + threadIdx.x;
  if (t >= Nn * Hd * PVd) return;
  int n = t / (Hd * PVd);
  int hp = t % (Hd * PVd);
  int h = hp / PVd, p = hp % PVd;
  const float* R = rot + n * 9;
  const float* T = trans + n * 3;
  float vx = opraw[(long)n * 288 + h * 24 + p * 3 + 0] - T[0];
  float vy = opraw[(long)n * 288 + h * 24 + p * 3 + 1] - T[1];
  float vz = opraw[(long)n * 288 + h * 24 + p * 3 + 2] - T[2];
  float s = 0.0f;
#pragma unroll
  for (int i = 0; i < 3; ++i) {
    // R^T: out_i = sum_j R[j][i] * v_j
    float li = R[0 * 3 + i] * vx + R[1 * 3 + i] * vy + R[2 * 3 + i] * vz;
    ocat[(long)n * CATd + 192 + 96 * i + hp] = li;
    s += li * li;
  }
  ocat[(long)n * CATd + 480 + hp] = sqrtf(s + 1e-8f);
}

// ---------------------------------------------------------------------------
static inline int cdiv(int a, int b) { return (a + b - 1) / b; }

extern "C" void kernel_launch(void* const* d_in, const int* in_sizes, int n_in,
                              void* d_out, int out_size, void* d_ws, size_t ws_size,
                              hipStream_t stream) {
  const float* s      = (const float*)d_in[0];
  const float* z      = (const float*)d_in[1];
  const float* rot    = (const float*)d_in[2];
  const float* trans  = (const float*)d_in[3];
  const float* mask   = (const float*)d_in[4];
  const float* w_q    = (const float*)d_in[5];
  const float* b_q    = (const float*)d_in[6];
  const float* w_kv   = (const float*)d_in[7];
  const float* b_kv   = (const float*)d_in[8];
  const float* w_qp   = (const float*)d_in[9];
  const float* b_qp   = (const float*)d_in[10];
  const float* w_kvp  = (const float*)d_in[11];
  const float* b_kvp  = (const float*)d_in[12];
  const float* w_b    = (const float*)d_in[13];
  const float* b_b    = (const float*)d_in[14];
  const float* head_w = (const float*)d_in[15];
  const float* w_out  = (const float*)d_in[16];
  const float* b_out  = (const float*)d_in[17];
  float* out = (float*)d_out;

  // Workspace layout (floats). Total ~38.3 MB.
  float* ws = (float*)d_ws;
  float* qbuf   = ws;                  // [512][192]
  float* kvbuf  = qbuf + 512 * 192;    // [512][384]
  float* qpraw  = kvbuf + 512 * 384;   // [512][144]
  float* kvpraw = qpraw + 512 * 144;   // [512][432]
  float* biasb  = kvpraw + 512 * 432;  // [512*512][12]
  float* qpts   = biasb + 512 * 512 * 12;  // [512][144]
  float* kptsT  = qpts + 512 * 144;    // [12*12][512]
  float* vpts   = kptsT + 144 * 512;   // [512][288]
  float* qn     = vpts + 512 * 288;    // [512][12]
  float* kn     = qn + 512 * 12;       // [512][12]
  float* kT     = kn + 512 * 12;       // [12*16][512]
  float* aTb    = kT + 192 * 512;      // [512][512][16]
  float* ocat   = aTb + 512 * 512 * 16;    // [512][2112]
  float* opraw  = ocat + 512 * 2112;   // [512][288]

  // 1) Projections from s (fp32 WMMA GEMMs, compile-time shapes)
  gemm_bias_kernel<192, 384, 384, 192, 192>
      <<<cdiv(32 * 12, 8), 256, 0, stream>>>(s, w_q, b_q, qbuf, 512);
  gemm_bias_kernel<384, 384, 384, 384, 384>
      <<<cdiv(32 * 24, 8), 256, 0, stream>>>(s, w_kv, b_kv, kvbuf, 512);
  gemm_bias_kernel<144, 384, 384, 144, 144>
      <<<cdiv(32 * 9, 8), 256, 0, stream>>>(s, w_qp, b_qp, qpraw, 512);
  gemm_bias_kernel<432, 384, 384, 432, 432>
      <<<cdiv(32 * 27, 8), 256, 0, stream>>>(s, w_kvp, b_kvp, kvpraw, 512);

  // 2) b_bias = z @ w_b + b_b  (streams z once from HBM; N=12 padded to 16)
  gemm_bias_kernel<12, 128, 128, 12, 12>
      <<<cdiv(16384, 8), 256, 0, stream>>>(z, w_b, b_b, biasb, 512 * 512);

  // 3) Frame transforms + layout packs
  pack_qpts_kernel<<<cdiv(Nn * Hd, 256), 256, 0, stream>>>(qpraw, rot, trans, qpts, qn);
  pack_kvpts_kernel<<<cdiv(Nn * Hd, 256), 256, 0, stream>>>(kvpraw, rot, trans,
                                                            kptsT, vpts, kn);
  pack_kT_kernel<<<cdiv(Nn * Hd * CHd, 256), 256, 0, stream>>>(kvbuf, kT);

  // 4) Fused attention: logits + softmax + o + o_pt, probs -> aT
  attn_kernel<<<Hd * (Nn / 16), 256, 0, stream>>>(qbuf, kT, kvbuf, qpts, kptsT,
                                                  vpts, qn, kn, biasb, mask,
                                                  head_w, aTb, ocat, opraw);

  // 5) o_pair (z hits L2 from step 2)
  opair_kernel<<<Nn, 256, 0, stream>>>(aTb, z, ocat);

  // 6) o_pt inverse rotation + norms into o_cat
  finalize_opt_kernel<<<cdiv(Nn * Hd * PVd, 256), 256, 0, stream>>>(opraw, rot,
                                                                    trans, ocat);

  // 7) Output projection: out = o_cat @ w_out + b_out
  gemm_bias_kernel<384, 2112, 2112, 384, 384>
      <<<cdiv(32 * 24, 8), 256, 0, stream>>>(ocat, w_out, b_out, out, 512);
}